// Model_67705864454710
// MI455X (gfx1250) — compile-verified
//
#include <hip/hip_runtime.h>

typedef __attribute__((ext_vector_type(2))) float v2f;
typedef __attribute__((ext_vector_type(8))) float v8f;

#define SPB        512
#define NCOEF      257
#define NPAD       272            // 257 padded to 17*16 (zero gains kill padding)
#define NPIX       1024           // 32*32 pixels
#define MT         32             // pixels per workgroup
#define ENV_STRIDE 516            // 512 + 4 pad -> conflict-free b64 LDS reads
#define PQ_STRIDE  276            // 272 + 4 pad -> conflict-free b64 LDS reads
#define LDS_FLOATS (MT*ENV_STRIDE + 2*MT*PQ_STRIDE)

__device__ __forceinline__ v8f wmma4(v2f a, v2f b, v8f c) {
  // D(16x16,f32) = A(16x4,f32) * B(4x16,f32) + C
  return __builtin_amdgcn_wmma_f32_16x16x4_f32(false, a, false, b, (short)0, c,
                                               false, false);
}

// Fill symmetric DFT tables: cosT[n*512+k] = cos(2*pi*n*k/512)/sqrt(512), sinT likewise.
__global__ void init_tables_kernel(float* __restrict__ cosT, float* __restrict__ sinT) {
  int idx = blockIdx.x * blockDim.x + threadIdx.x;   // 1024 blocks * 256 = 262144
  int n = idx >> 9, k = idx & 511;
  int ph = (n * k) & 511;                            // exact mod-512 phase
  float ang = (float)ph * (6.28318530717958647692f / 512.0f);
  float s, c;
  __sincosf(ang, &s, &c);
  const float inv = 0.04419417382415922f;            // 1/sqrt(512)
  cosT[idx] = c * inv;
  sinT[idx] = s * inv;
}

__global__ void zero_kernel(float* __restrict__ p, int nElem) {
  int i = blockIdx.x * blockDim.x + threadIdx.x;
  if (i < nElem) p[i] = 0.0f;
}

// One scan step: sf_mid = sf_in + irfft(g * rfft(env)), mic = sf_mid[16,16].
__global__ void __launch_bounds__(256)
step_kernel(const float* __restrict__ sf_in, float* __restrict__ sf_mid,
            const float* __restrict__ imp, const float* __restrict__ tf,
            const float* __restrict__ nz, const float* __restrict__ cosT,
            const float* __restrict__ sinT, float* __restrict__ mic) {
  extern __shared__ float lds[];
  float* envL = lds;                      // MT x ENV_STRIDE
  float* Pl   = lds + MT * ENV_STRIDE;    // MT x PQ_STRIDE
  float* Ql   = Pl  + MT * PQ_STRIDE;     // MT x PQ_STRIDE

  const int tid  = threadIdx.x;
  const int wave = tid >> 5;
  const int lane = tid & 31;
  const int lo   = lane & 15;
  const int hi   = lane >> 4;
  const int pixBase = blockIdx.x * MT;

  // ---- Phase 0: envelope = clip(lin_resize16->512(imp),0,1) * noise, into LDS ----
  for (int e = tid; e < MT * SPB; e += 256) {
    int p = e >> 9, n = e & 511;
    float c  = (float)n * 0.03125f - 0.484375f;      // (n+0.5)/32 - 0.5
    float fl = floorf(c);
    int   i0 = (int)fl;
    float f  = c - fl;
    int   i1 = i0 + 1;
    i0 = i0 < 0 ? 0 : (i0 > 15 ? 15 : i0);
    i1 = i1 < 0 ? 0 : (i1 > 15 ? 15 : i1);
    const float* ip = imp + (size_t)(pixBase + p) * 16;
    float env = ip[i0] * (1.0f - f) + ip[i1] * f;
    env = fminf(fmaxf(env, 0.0f), 1.0f);
    env *= nz[(size_t)(pixBase + p) * SPB + n];
    envL[p * ENV_STRIDE + n] = env;
  }
  __syncthreads();

  // ---- Phase 1: P = env*CT, Q = env*ST  (M=32, K=512, N=272), then scale by w*g ----
  for (int job = wave; job < 2 * (NPAD / 16); job += 8) {
    const int m0 = (job & 1) * 16;
    const int n0 = (job >> 1) * 16;
    v8f cp = {};
    v8f cq = {};
    const float* envRow = envL + (m0 + lo) * ENV_STRIDE + hi * 2;
    for (int k = 0; k < SPB; k += 4) {
      v2f a = *(const v2f*)(envRow + k);             // A frag: rows lo, K = k+2*hi..+1
      const int kk = k + hi * 2;
      const float* cB = cosT + (size_t)kk * 512 + (n0 + lo);
      const float* sB = sinT + (size_t)kk * 512 + (n0 + lo);
      v2f bc; bc.x = cB[0]; bc.y = cB[512];
      v2f bs; bs.x = sB[0]; bs.y = sB[512];
      cp = wmma4(a, bc, cp);
      cq = wmma4(a, bs, cq);
    }
    const int  coeff = n0 + lo;
    const float w    = (coeff == 0 || coeff == 256) ? 1.0f : 2.0f;
    const bool valid = (coeff < NCOEF);
#pragma unroll
    for (int v = 0; v < 8; ++v) {
      int p = m0 + v + 8 * hi;
      float g = 0.0f;
      if (valid) {
        float t = tf[(size_t)(pixBase + p) * NCOEF + coeff];
        g = w * fminf(fmaxf(t, 0.0f), 1.0f);
      }
      Pl[p * PQ_STRIDE + coeff] = cp[v] * g;
      Ql[p * PQ_STRIDE + coeff] = cq[v] * g;
    }
  }
  __syncthreads();

  // ---- Phase 2: out = sf_in + P'*CT + Q'*ST  (M=32, K=272, N=512) ----
  for (int job = wave; job < 2 * (SPB / 16); job += 8) {
    const int m0 = (job & 1) * 16;
    const int n0 = (job >> 1) * 16;
    v8f c;
#pragma unroll
    for (int v = 0; v < 8; ++v) {
      int p = m0 + v + 8 * hi;
      c[v] = sf_in[(size_t)(pixBase + p) * SPB + (n0 + lo)];
    }
    const float* pRow = Pl + (m0 + lo) * PQ_STRIDE + hi * 2;
    const float* qRow = Ql + (m0 + lo) * PQ_STRIDE + hi * 2;
    for (int k = 0; k < NPAD; k += 4) {
      v2f ap = *(const v2f*)(pRow + k);
      v2f aq = *(const v2f*)(qRow + k);
      const int kk = k + hi * 2;
      const float* cB = cosT + (size_t)kk * 512 + (n0 + lo);
      const float* sB = sinT + (size_t)kk * 512 + (n0 + lo);
      v2f bc; bc.x = cB[0]; bc.y = cB[512];
      v2f bs; bs.x = sB[0]; bs.y = sB[512];
      c = wmma4(ap, bc, c);
      c = wmma4(aq, bs, c);
    }
#pragma unroll
    for (int v = 0; v < 8; ++v) {
      int p  = m0 + v + 8 * hi;
      int pg = pixBase + p;
      float out = c[v];
      sf_mid[(size_t)pg * SPB + (n0 + lo)] = out;
      if (pg == 16 * 32 + 16) mic[n0 + lo] = out;     // mic pixel (16,16)
    }
  }
}

// 3x3 reflect-padded box mean over the 32x32 grid, per channel.
__global__ void blur_kernel(const float* __restrict__ sf_mid, float* __restrict__ sf_out) {
  const int pix = blockIdx.x;
  const int h = pix >> 5, w = pix & 31;
  for (int n = threadIdx.x; n < SPB; n += blockDim.x) {
    float acc = 0.0f;
#pragma unroll
    for (int dh = -1; dh <= 1; ++dh) {
      int hh = h + dh;
      hh = hh < 0 ? -hh : (hh > 31 ? 62 - hh : hh);
#pragma unroll
      for (int dw = -1; dw <= 1; ++dw) {
        int ww = w + dw;
        ww = ww < 0 ? -ww : (ww > 31 ? 62 - ww : ww);
        acc += sf_mid[(size_t)(hh * 32 + ww) * SPB + n];
      }
    }
    sf_out[(size_t)pix * SPB + n] = acc * (1.0f / 9.0f);
  }
}

extern "C" void kernel_launch(void* const* d_in, const int* in_sizes, int n_in,
                              void* d_out, int out_size, void* d_ws, size_t ws_size,
                              hipStream_t stream) {
  // inputs: [0]=x (unused), [1]=transfer_functions, [2]=impulses, [3]=noise
  const float* tf  = (const float*)d_in[1];
  const float* imp = (const float*)d_in[2];
  const float* nz  = (const float*)d_in[3];
  float*       mic = (float*)d_out;

  float* ws   = (float*)d_ws;
  float* cosT = ws;                       // 512*512
  float* sinT = cosT + 512 * 512;         // 512*512
  float* sfA  = sinT + 512 * 512;         // 1024*512 state (post-blur)
  float* sfB  = sfA  + NPIX * SPB;        // 1024*512 pre-blur

  init_tables_kernel<<<1024, 256, 0, stream>>>(cosT, sinT);
  zero_kernel<<<(NPIX * SPB + 255) / 256, 256, 0, stream>>>(sfA, NPIX * SPB);

  const size_t ldsBytes = (size_t)LDS_FLOATS * sizeof(float);
  for (int t = 0; t < 128; ++t) {
    step_kernel<<<NPIX / MT, 256, ldsBytes, stream>>>(
        sfA, sfB,
        imp + (size_t)t * NPIX * 16,
        tf  + (size_t)t * NPIX * NCOEF,
        nz  + (size_t)t * NPIX * SPB,
        cosT, sinT,
        mic + (size_t)t * SPB);
    blur_kernel<<<NPIX, 256, 0, stream>>>(sfB, sfA);
  }
}